// SpanMaxPooler_60748017435289
// MI455X (gfx1250) — compile-verified
//
#include <hip/hip_runtime.h>
#include <cfloat>
#include <cstdint>

// Problem shape (fixed by the reference): B=64, S=512, H=1024, I=2
#define NB 64
#define NS 512
#define NH 1024
#define NI 2
#define THREADS 256          // 8 wave32s per workgroup
#define MAX_SPAN 32          // reference guarantees span length 1..32

typedef uint32_t v4u __attribute__((ext_vector_type(4)));
typedef uint32_t v8u __attribute__((ext_vector_type(8)));

// One workgroup per (b, i) span. The span fetch is a 2D tile copy
// (L rows x 4KB from the strided [S, H] tensor) -> exactly one CDNA5 Tensor
// Data Mover descriptor: tensor_load_to_lds DMAs the whole span tile into LDS
// (TENSORcnt), wave 0 waits on s_wait_tensorcnt, one workgroup barrier
// publishes the tile, then all 8 waves max-reduce their H-slices from LDS.
__global__ __launch_bounds__(THREADS) void span_max_pool_kernel(
    const float* __restrict__ hidden,   // [B, S, H]
    const int*   __restrict__ starts,   // [B, I]
    const int*   __restrict__ ends,     // [B, I]
    const float* __restrict__ miss,     // [I, H]
    float*       __restrict__ out)      // [B, I*H]
{
  __shared__ float4 slab[MAX_SPAN * THREADS];  // 128 KB (of 320 KB/WGP pool)

  const int blk = blockIdx.x;                // 0 .. B*I-1
  const int b   = blk / NI;
  const int i   = blk - b * NI;
  const int tid = threadIdx.x;               // 0 .. 255

  const int s0 = starts[b * NI + i];
  const int e0 = ends[b * NI + i];

  float4 acc;
  if (s0 < 0 || e0 < 0) {
    // Missing span: emit the learned fallback embedding (block-uniform branch).
    acc = reinterpret_cast<const float4*>(miss)[i * (NH / 4) + tid];
  } else {
    acc = make_float4(-FLT_MAX, -FLT_MAX, -FLT_MAX, -FLT_MAX);
    const int L = e0 - s0;                   // 1 .. 32

    if (tid < 32) {  // wave 0 drives the TDM engine (D# lives in SGPRs)
      // LDS byte address of slab base (flat-pointer low 32 bits == LDS offset).
      const uint32_t lds0 = (uint32_t)(uintptr_t)(&slab[0]);
      const uint64_t ga   = (uint64_t)(uintptr_t)(
          hidden + ((size_t)b * NS + (size_t)s0) * NH);

      // ---- D# group 0 (4 SGPRs): control / lds_addr / global_addr / type
      v4u g0;
      g0[0] = 1u;                                   // count=1, user mode
      g0[1] = lds0;                                 // lds_addr[31:0]
      g0[2] = (uint32_t)ga;                         // global_addr[31:0]
      g0[3] = ((uint32_t)(ga >> 32) & 0x01FFFFFFu)  // global_addr[56:32]
            | 0x80000000u;                          // type=2 ("image")

      // ---- D# group 1 (8 SGPRs): dims / tile / strides (2D tile, L x 1024)
      v8u g1;
      g1[0] = 2u << 16;                             // data_size=2 (4 bytes)
      g1[1] = ((uint32_t)NH & 0xFFFFu) << 16;       // tensor_dim0[15:0]=1024
      g1[2] = ((uint32_t)L & 0xFFFFu) << 16;        // dim0[31:16]=0 | tensor_dim1[15:0]=L
      g1[3] = (uint32_t)NH << 16;                   // tensor_dim1[31:16]=0 | tile_dim0=1024
      g1[4] = (uint32_t)L & 0xFFFFu;                // tile_dim1=L | tile_dim2=0
      g1[5] = (uint32_t)NH;                         // tensor_dim0_stride[31:0]=1024
      g1[6] = 0u;                                   // stride0[47:32]=0 | stride1[15:0]=0
      g1[7] = 0u;                                   // tensor_dim1_stride[47:16]=0

      // 2D tensor: VADDR2/VADDR3 are NULL -> 2-group form. One DMA moves the
      // entire span tile (up to 32 x 4 KB) into LDS.
      asm volatile("tensor_load_to_lds %0, %1"
                   :: "s"(g0), "s"(g1)
                   : "memory");
      // Drain the DMA (per-wave TENSORcnt; only wave 0 issued).
      __builtin_amdgcn_s_wait_tensorcnt(0);
    }
    // Publish the LDS tile to all 8 waves.
    __syncthreads();

    // Consume: ds_load_b128 + component-wise max. Lane `tid` owns floats
    // [tid*4, tid*4+4) of H; tile row r sits at slab bytes [r*4096, ...).
    for (int r = 0; r < L; ++r) {
      const float4 v = slab[r * THREADS + tid];
      acc.x = fmaxf(acc.x, v.x);
      acc.y = fmaxf(acc.y, v.y);
      acc.z = fmaxf(acc.z, v.z);
      acc.w = fmaxf(acc.w, v.w);
    }
  }

  // out[b, i*H + tid*4 .. +4)
  reinterpret_cast<float4*>(out)[(size_t)b * (NI * NH / 4) + i * (NH / 4) + tid] = acc;
}

extern "C" void kernel_launch(void* const* d_in, const int* in_sizes, int n_in,
                              void* d_out, int out_size, void* d_ws, size_t ws_size,
                              hipStream_t stream) {
  (void)in_sizes; (void)n_in; (void)out_size; (void)d_ws; (void)ws_size;
  const float* hidden = (const float*)d_in[0];
  const int*   starts = (const int*)d_in[1];
  const int*   ends   = (const int*)d_in[2];
  const float* miss   = (const float*)d_in[3];
  float*       out    = (float*)d_out;

  span_max_pool_kernel<<<dim3(NB * NI), dim3(THREADS), 0, stream>>>(
      hidden, starts, ends, miss, out);
}